// BBoxPrior_18769007083638
// MI455X (gfx1250) — compile-verified
//
#include <hip/hip_runtime.h>
#include <hip/hip_bf16.h>

#define MAXN 128  // reference uses N=100 GT boxes per image

typedef unsigned u32;
typedef u32 u32x4 __attribute__((ext_vector_type(4)));
typedef int  i32x8 __attribute__((ext_vector_type(8)));

// ---------------------------------------------------------------------------
// CDNA5 async global->LDS copies (gfx1250 ASYNCcnt-tracked path).
// Flat shared pointers on gfx1250 are {LDS_aperture[63:32], lds_offset[31:0]},
// so truncation to 32 bits yields the LDS byte offset the VDST operand wants.
// ---------------------------------------------------------------------------
__device__ __forceinline__ unsigned lds_off(const void* p) {
  return (unsigned)(unsigned long long)(size_t)p;
}
__device__ __forceinline__ void async_ld_b64(void* lds_dst, const void* g_src) {
  asm volatile("global_load_async_to_lds_b64 %0, %1, off"
               :: "v"(lds_off(lds_dst)), "v"(g_src) : "memory");
}
__device__ __forceinline__ void async_ld_b32(void* lds_dst, const void* g_src) {
  asm volatile("global_load_async_to_lds_b32 %0, %1, off"
               :: "v"(lds_off(lds_dst)), "v"(g_src) : "memory");
}
__device__ __forceinline__ void wait_async0() {
  asm volatile("s_wait_asynccnt 0" ::: "memory");
}

// ---------------------------------------------------------------------------
// Tensor Data Mover: 1-D dword copy global -> LDS via D# groups 0-1 only
// (VADDR2/VADDR3 omitted => <=2-D tensor mode). Per CDNA5 ISA:
//   group0: [1:0]=count=1, [63:32]=lds_addr, [120:64]=global_addr, [127:126]=2
//   group1: [17:16]=data_size(2 => 4B), [79:48]=tensor_dim0,
//           [127:112]=tile_dim0, dims 1/2 unused, stride0=tensor_dim0
// Issue from ONE wave only (TDM ignores EXEC; it is a per-wave DMA command).
// ---------------------------------------------------------------------------
__device__ __forceinline__ void tdm_load_dwords(void* lds_dst, const void* g_src,
                                                int ndwords) {
  unsigned long long ga = (unsigned long long)(size_t)g_src;
  u32x4 g0;
  g0[0] = 1u;                                            // count=1, user mode
  g0[1] = lds_off(lds_dst);                              // lds_addr
  g0[2] = (u32)ga;                                       // global_addr[31:0]
  g0[3] = (u32)((ga >> 32) & 0x01FFFFFFu) | (2u << 30);  // addr[56:32], type=2
  i32x8 g1;
  g1[0] = (2 << 16);                                     // data_size = 4 bytes
  g1[1] = (ndwords & 0xFFFF) << 16;                      // tensor_dim0[15:0]
  g1[2] = ((ndwords >> 16) & 0xFFFF) | (1 << 16);        // dim0[31:16], dim1=1
  g1[3] = (ndwords & 0xFFFF) << 16;                      // tile_dim0 = ndwords
  g1[4] = 0;                                             // tile_dim1/2 unused
  g1[5] = ndwords;                                       // tensor_dim0_stride
  g1[6] = 0;
  g1[7] = 0;
  asm volatile("tensor_load_to_lds %0, %1" :: "s"(g0), "s"(g1) : "memory");
}

// Pack (iou, anchor) so that u64-max == (max iou, first-occurrence argmax).
// IoU >= 0, so its float bits are monotone as unsigned. ~anchor in the low
// word makes ties resolve to the SMALLEST anchor index (jnp.argmax semantics).
__device__ __forceinline__ unsigned long long pack_iou(float iou, unsigned a) {
  return ((unsigned long long)__float_as_uint(iou) << 32) |
         (unsigned long long)(~a);
}
__device__ __forceinline__ float unpack_iou(unsigned long long p) {
  return __uint_as_float((unsigned)(p >> 32));
}
__device__ __forceinline__ unsigned unpack_anchor(unsigned long long p) {
  return ~(unsigned)(p & 0xffffffffULL);
}

// ---------------------------------------------------------------------------
// K1: per (batch, gt-box) max/argmax of IoU over all anchors.
// grid = (N, B), block = 256. Packed result -> workspace [B*N] u64.
// Anchors stream coalesced (L2-resident, 1 MB); no sharing per thread, so
// plain b128 loads are the right path here.
// ---------------------------------------------------------------------------
__global__ void __launch_bounds__(256)
k_box_best(const float4* __restrict__ anchors,       // [A] cbox (cx,cy,w,h)
           const float4* __restrict__ bboxes,        // [B*N] bbox (x1,y1,x2,y2)
           unsigned long long* __restrict__ boxbest, // [B*N] packed
           int A, int N) {
  const int n = blockIdx.x, b = blockIdx.y, tid = threadIdx.x;
  const float4 box = bboxes[(size_t)b * N + n];
  const float areaB = (box.z - box.x) * (box.w - box.y);

  unsigned long long best = 0ULL;
  for (int a = tid; a < A; a += 256) {
    float4 c = anchors[a];
    float ax1 = c.x - 0.5f * c.z, ay1 = c.y - 0.5f * c.w;
    float ax2 = c.x + 0.5f * c.z, ay2 = c.y + 0.5f * c.w;
    float iw = fmaxf(fminf(ax2, box.z) - fmaxf(ax1, box.x), 0.0f);
    float ih = fmaxf(fminf(ay2, box.w) - fmaxf(ay1, box.y), 0.0f);
    float inter = iw * ih;
    float iou = inter / (c.z * c.w + areaB - inter);
    unsigned long long p = pack_iou(iou, (unsigned)a);
    best = (p > best) ? p : best;
  }

  __shared__ unsigned long long red[256];
  red[tid] = best;
  __syncthreads();
  for (int s = 128; s > 0; s >>= 1) {
    if (tid < s) {
      unsigned long long o = red[tid + s];
      if (o > red[tid]) red[tid] = o;
    }
    __syncthreads();
  }
  if (tid == 0) boxbest[(size_t)b * N + n] = red[0];
}

// ---------------------------------------------------------------------------
// K2: per-anchor matching + score + delta encoding.
// grid = (ceil(A/256), B), block = 256.
// Broadcast-reused data staged in LDS: GT boxes via the Tensor Data Mover
// (TENSORcnt), packed box-maxima + labels via async-to-LDS (ASYNCcnt).
// ---------------------------------------------------------------------------
__global__ void __launch_bounds__(256)
k_finalize(const float4* __restrict__ anchors,       // [A] cbox
           const int* __restrict__ labels,           // [B*N]
           const float4* __restrict__ bboxes,        // [B*N] bbox
           const float4* __restrict__ enc_mean,      // [4]
           const float4* __restrict__ enc_std,       // [4]
           const float* __restrict__ iou_thr_p,      // scalar
           const unsigned long long* __restrict__ boxbest, // [B*N]
           float* __restrict__ out,                  // conf [B*A*C] ++ deltas [B*A*4]
           int A, int N, int B, int C) {
  const int tid = threadIdx.x;
  const int b = blockIdx.y;
  const int a = blockIdx.x * 256 + tid;

  __shared__ float4 sBox[MAXN];
  __shared__ unsigned long long sBest[MAXN];
  __shared__ int sLabel[MAXN];

  // Wave 0: one TDM descriptor moves the whole N*16-byte GT-box tile.
  if (tid < 32) {
    tdm_load_dwords(&sBox[0], bboxes + (size_t)b * N, N * 4);
  }
  // Waves 0..3 (lanes t<N): async-to-LDS for packed maxima + labels.
  for (int t = tid; t < N; t += 256) {
    async_ld_b64(&sBest[t], boxbest + (size_t)b * N + t);
    async_ld_b32(&sLabel[t], labels + (size_t)b * N + t);
  }
  if (a < A) __builtin_prefetch(anchors + a, 0, 0);   // global_prefetch_b8

  wait_async0();                                      // s_wait_asynccnt 0
  __builtin_amdgcn_s_wait_tensorcnt(0);               // s_wait_tensorcnt 0
  __syncthreads();                                    // publish LDS to 8 waves
  if (a >= A) return;

  const float thr = *iou_thr_p;
  const float4 em = *enc_mean;
  const float4 es = *enc_std;

  const float4 c = anchors[a];
  const float ax1 = c.x - 0.5f * c.z, ay1 = c.y - 0.5f * c.w;
  const float ax2 = c.x + 0.5f * c.z, ay2 = c.y + 0.5f * c.w;
  const float areaA = c.z * c.w;

  float bestIou = -1.0f;
  int bestN = 0;
  int winner = -1;  // segment_max of box indices mapping to this anchor
  for (int n = 0; n < N; ++n) {
    float4 bb = sBox[n];
    float iw = fmaxf(fminf(ax2, bb.z) - fmaxf(ax1, bb.x), 0.0f);
    float ih = fmaxf(fminf(ay2, bb.w) - fmaxf(ay1, bb.y), 0.0f);
    float inter = iw * ih;
    float areaB = (bb.z - bb.x) * (bb.w - bb.y);
    float iou = inter / (areaA + areaB - inter);
    if (iou > bestIou) { bestIou = iou; bestN = n; }         // first-max argmax
    if (unpack_anchor(sBest[n]) == (unsigned)a) winner = n;  // largest n wins
  }

  int box_idx = bestN;
  float miou = bestIou;
  if (winner >= 0) {
    box_idx = winner;
    miou = unpack_iou(sBest[winner]);
  }
  float denom = fmaxf(unpack_iou(sBest[box_idx]), thr);
  if (miou < 0.5f * thr) miou = 0.0f;
  float score = miou / denom;
  int lab = sLabel[box_idx];
  if (lab <= 0) { score = 0.0f; lab = 0; }

  const size_t row = (size_t)b * A + a;
  for (int cc = 0; cc < C; ++cc)
    out[row * C + cc] = (lab == cc + 1) ? score : 0.0f;

  // delta encoding: matched bbox -> cbox -> (d - mean) / std
  float4 m = sBox[box_idx];
  float mcx = 0.5f * (m.x + m.z), mcy = 0.5f * (m.y + m.w);
  float mw = m.z - m.x, mh = m.w - m.y;
  float d0 = ((mcx - c.x) / c.z - em.x) / es.x;
  float d1 = ((mcy - c.y) / c.w - em.y) / es.y;
  float d2 = (logf(mw / c.z) - em.z) / es.z;
  float d3 = (logf(mh / c.w) - em.w) / es.w;
  // deltas live after the conf block; B*A is a multiple of 4 -> 16B aligned
  float4* dout = (float4*)(out + (size_t)B * A * C) + row;
  *dout = make_float4(d0, d1, d2, d3);
}

// ---------------------------------------------------------------------------
extern "C" void kernel_launch(void* const* d_in, const int* in_sizes, int n_in,
                              void* d_out, int out_size, void* d_ws, size_t ws_size,
                              hipStream_t stream) {
  const float* anchors  = (const float*)d_in[0];   // [A,4] fp32 cbox
  const int*   labels   = (const int*)d_in[1];     // [B,N] int32
  const float* bboxes   = (const float*)d_in[2];   // [B,N,4] fp32 bbox
  const float* enc_mean = (const float*)d_in[3];   // [4]
  const float* enc_std  = (const float*)d_in[4];   // [4]
  const float* iou_thr  = (const float*)d_in[5];   // scalar
  // d_in[6] = num_classes; derived from out_size instead (host-side index math)

  const int A = in_sizes[0] / 4;
  const int N = 100;                 // reference constant (gt boxes per image)
  const int B = in_sizes[1] / N;
  const int C = out_size / (B * A) - 4;   // out = B*A*C conf + B*A*4 deltas

  unsigned long long* boxbest = (unsigned long long*)d_ws;  // [B*N] u64

  k_box_best<<<dim3(N, B), 256, 0, stream>>>(
      (const float4*)anchors, (const float4*)bboxes, boxbest, A, N);

  const int gx = (A + 255) / 256;
  k_finalize<<<dim3(gx, B), 256, 0, stream>>>(
      (const float4*)anchors, labels, (const float4*)bboxes,
      (const float4*)enc_mean, (const float4*)enc_std, iou_thr,
      boxbest, (float*)d_out, A, N, B, C);
}